// SinkhornTransportModel_62826781605906
// MI455X (gfx1250) — compile-verified
//
#include <hip/hip_runtime.h>

#define KV 64
#define THRESH 5e-07f
#define EPSF 1e-08f
#define MAX_ITERS 1500
#define M_STRIDE 72   // padded row stride (floats) for 64x64 kernel matrix in LDS
#define P_STRIDE 68   // padded row stride (floats) for per-pixel K-vectors in LDS
#define PPB 16        // pixels per block (one wave32 per block)

typedef __attribute__((ext_vector_type(2))) float v2f;
typedef __attribute__((ext_vector_type(8))) float v8f;

__global__ __launch_bounds__(32)
void sinkhorn_posterior_kernel(const float* __restrict__ u_s,
                               const float* __restrict__ u_t,
                               const int*   __restrict__ z_t,
                               const float* __restrict__ Cm,
                               float* __restrict__ out)
{
    // LDS: M (64x72) + U,V (16x68 each) = 6784 floats = 27136 B
    __shared__ float smem[KV * M_STRIDE + 2 * PPB * P_STRIDE];
    float* Mk = smem;                   // kernel matrix exp(-C)
    float* Ul = Mk + KV * M_STRIDE;     // row scaling u (LDS copy for A-frag reads)
    float* Vl = Ul + PPB * P_STRIDE;    // col scaling v (LDS copy for A-frag reads)

    const int lane    = threadIdx.x;
    const int pixBase = blockIdx.x * PPB;
    const int p16     = lane & 15;
    const int hi      = lane >> 4;

    // ---- init kernel matrix M = exp(-C / LAM), LAM == 1 ----
    for (int idx = lane; idx < KV * KV; idx += 32) {
        int i = idx >> 6;
        int j = idx & 63;
        Mk[i * M_STRIDE + j] = __expf(-Cm[idx]);
    }

    // ---- init u = v = 1 in LDS (first-iteration A-fragments) ----
    {
        float* lu = Ul + p16 * P_STRIDE + hi * 32;
        float* lv = Vl + p16 * P_STRIDE + hi * 32;
        const float4 one4 = make_float4(1.f, 1.f, 1.f, 1.f);
        #pragma unroll
        for (int q = 0; q < 8; ++q) {
            *(float4*)(lu + 4 * q) = one4;
            *(float4*)(lv + 4 * q) = one4;
        }
    }
    __syncthreads();

    // ---- hoist loop-invariant WMMA B-fragments of M into registers ----
    // fT[c][t]: fragment of M^T  (element [k][n] = M[n][k], contiguous in M rows)
    // fM[c][t]: fragment of M    (element [k][n] = M[k][n])
    v2f fT[16][4];
    v2f fM[16][4];
    #pragma unroll
    for (int c = 0; c < 16; ++c) {
        const int kb = 4 * c + 2 * hi;
        #pragma unroll
        for (int t = 0; t < 4; ++t) {
            fT[c][t] = *(const v2f*)(Mk + (16 * t + p16) * M_STRIDE + kb);
            v2f m;
            m.x = Mk[(kb + 0) * M_STRIDE + 16 * t + p16];
            m.y = Mk[(kb + 1) * M_STRIDE + 16 * t + p16];
            fM[c][t] = m;
        }
    }

    // ---- marginals and scaling vectors in C/D-layout registers ----
    // element [t][vg] lives at pixel = vg + 8*hi, vocab col = 16*t + p16
    float aCD[4][8], bCD[4][8], uCD[4][8], vCD[4][8];
    #pragma unroll
    for (int t = 0; t < 4; ++t) {
        const int col = 16 * t + p16;
        #pragma unroll
        for (int vg = 0; vg < 8; ++vg) {
            const size_t g = (size_t)(pixBase + vg + 8 * hi) * KV + col;
            aCD[t][vg] = u_s[g];
            bCD[t][vg] = u_t[g];
            uCD[t][vg] = 1.f;
            vCD[t][vg] = 1.f;
        }
    }

    // ---- Sinkhorn iterations in u/v form, fp32 WMMA matvecs ----
    float err  = 1.0f;
    int   iter = 0;
    while (err > THRESH && iter < MAX_ITERS) {
        float lerr = 0.0f;

        // step 1: r[p][i] = sum_j V[p][j] * M[i][j]
        v8f acc[4] = {};
        #pragma unroll
        for (int c = 0; c < 16; ++c) {
            v2f av = *(const v2f*)(Vl + p16 * P_STRIDE + 4 * c + 2 * hi);
            #pragma unroll
            for (int t = 0; t < 4; ++t)
                acc[t] = __builtin_amdgcn_wmma_f32_16x16x4_f32(
                    false, av, false, fT[c][t], (short)0, acc[t], false, false);
        }
        // u <- a / (r + eps); row-marginal error (pre-update u)
        #pragma unroll
        for (int t = 0; t < 4; ++t) {
            const int col = 16 * t + p16;
            #pragma unroll
            for (int vg = 0; vg < 8; ++vg) {
                float r  = acc[t][vg];
                lerr = fmaxf(lerr, fabsf(uCD[t][vg] * r - aCD[t][vg]));
                float un = aCD[t][vg] / (r + EPSF);
                uCD[t][vg] = un;
                Ul[(vg + 8 * hi) * P_STRIDE + col] = un;
            }
        }

        // step 2: s[p][j] = sum_i U[p][i] * M[i][j]
        v8f acc2[4] = {};
        #pragma unroll
        for (int c = 0; c < 16; ++c) {
            v2f av = *(const v2f*)(Ul + p16 * P_STRIDE + 4 * c + 2 * hi);
            #pragma unroll
            for (int t = 0; t < 4; ++t)
                acc2[t] = __builtin_amdgcn_wmma_f32_16x16x4_f32(
                    false, av, false, fM[c][t], (short)0, acc2[t], false, false);
        }
        // v <- b / (s + eps); column-marginal error
        #pragma unroll
        for (int t = 0; t < 4; ++t) {
            const int col = 16 * t + p16;
            #pragma unroll
            for (int vg = 0; vg < 8; ++vg) {
                float s  = acc2[t][vg];
                lerr = fmaxf(lerr, fabsf(vCD[t][vg] * s - bCD[t][vg]));
                float vn = bCD[t][vg] / (s + EPSF);
                vCD[t][vg] = vn;
                Vl[(vg + 8 * hi) * P_STRIDE + col] = vn;
            }
        }

        // wave-wide max error (uniform branch keeps EXEC all-1s for WMMA)
        lerr = fmaxf(lerr, __shfl_xor(lerr, 16));
        lerr = fmaxf(lerr, __shfl_xor(lerr, 8));
        lerr = fmaxf(lerr, __shfl_xor(lerr, 4));
        lerr = fmaxf(lerr, __shfl_xor(lerr, 2));
        lerr = fmaxf(lerr, __shfl_xor(lerr, 1));
        err = lerr;
        ++iter;
    }

    // ---- posterior column z: out[p][i] = u_i M[i][z] v_z / (sum_i u_i M[i][z] v_z + eps) ----
    {
        const int   z   = z_t[pixBase + p16];
        const float vz  = Vl[p16 * P_STRIDE + z];
        const float* ur = Ul + p16 * P_STRIDE + 32 * hi;
        float s = 0.f;
        #pragma unroll
        for (int j = 0; j < 32; ++j)
            s += ur[j] * Mk[(32 * hi + j) * M_STRIDE + z];
        s += __shfl_xor(s, 16);                 // combine the two half-rows per pixel
        const float scale = vz / (s * vz + EPSF);
        float* op = out + (size_t)(pixBase + p16) * KV + 32 * hi;
        #pragma unroll
        for (int j = 0; j < 32; j += 4) {
            float4 o4;
            o4.x = ur[j + 0] * Mk[(32 * hi + j + 0) * M_STRIDE + z] * scale;
            o4.y = ur[j + 1] * Mk[(32 * hi + j + 1) * M_STRIDE + z] * scale;
            o4.z = ur[j + 2] * Mk[(32 * hi + j + 2) * M_STRIDE + z] * scale;
            o4.w = ur[j + 3] * Mk[(32 * hi + j + 3) * M_STRIDE + z] * scale;
            *(float4*)(op + j) = o4;
        }
    }
}

extern "C" void kernel_launch(void* const* d_in, const int* in_sizes, int n_in,
                              void* d_out, int out_size, void* d_ws, size_t ws_size,
                              hipStream_t stream) {
    const float* u_s = (const float*)d_in[0];
    const float* u_t = (const float*)d_in[1];
    const int*   zt  = (const int*)d_in[2];
    const float* Cm  = (const float*)d_in[3];
    float* out       = (float*)d_out;

    const int npix   = in_sizes[2];               // B*H*W = 4096 pixels
    const int blocks = (npix + PPB - 1) / PPB;    // 256 blocks, 1 wave32 each
    sinkhorn_posterior_kernel<<<blocks, 32, 0, stream>>>(u_s, u_t, zt, Cm, out);
}